// S4Backbone_59493886984218
// MI455X (gfx1250) — compile-verified
//
#include <hip/hip_runtime.h>
#include <hip/hip_bf16.h>
#include <math.h>
#include <stdint.h>

// ---------------------------------------------------------------------------
// S4 backbone forward for MI455X (gfx1250, wave32).
// GEMMs: V_WMMA_F32_16X16X4_F32 (fp32 matrix pipe, full reference precision).
// A and B operands of channel GEMMs staged in LDS by the Tensor Data Mover
// (tensor_load_to_lds, double-buffered B, TENSORcnt-synchronized).
// S4 conv: per-row 2048-pt radix-2 FFT in LDS; kernel spectrum precomputed.
// ---------------------------------------------------------------------------

#define Bn    16
#define Ln    1024
#define DIN   64
#define Hn    256
#define Nn    64
#define TEMB  128
#define NBLK  6
#define NFFT  2048

typedef float v2f __attribute__((ext_vector_type(2)));
typedef float v8f __attribute__((ext_vector_type(8)));
typedef unsigned int u32x4 __attribute__((ext_vector_type(4)));
typedef int i32x4 __attribute__((ext_vector_type(4)));
typedef int i32x8 __attribute__((ext_vector_type(8)));

// Gate ONLY on builtin availability; clang major picks the builtin ARITY
// (ROCm 7.2 / clang-22: 5 args; upstream clang-23+: 6 args).
#if __has_builtin(__builtin_amdgcn_tensor_load_to_lds) && \
    __has_builtin(__builtin_amdgcn_s_wait_tensorcnt)
#define USE_TDM 1
#else
#define USE_TDM 0
#endif

#if USE_TDM
#if defined(__clang_major__) && (__clang_major__ >= 23)
#warning "CDNA5 TDM path ENABLED (6-arg tensor_load_to_lds, clang>=23)"
#else
#warning "CDNA5 TDM path ENABLED (5-arg tensor_load_to_lds, clang<23)"
#endif
#else
#warning "CDNA5 TDM path DISABLED: builtins not available on this pass"
#endif

// B chunk staged per K-step: 32 rows x 64 cols, padded to 72-float row stride
// (TDM pad: 8 DWORDs after every 64-DWORD row -> 2*72 = 16 mod 64 banks,
//  so the k / k+2 reads of the two half-waves hit disjoint bank groups).
#define BROW 72
#define KCH  32
#define NCH  (Hn / KCH)

#if USE_TDM
// Issue a 2D TDM load: tile (tile1 rows x tile0 fp32 elems), global row stride
// stride0 elems, into LDS at lds_off; pad 'pada' code DWORDs after every
// 'padi' code interval (D# pad_interval / pad_amount encodings).
__device__ __forceinline__ void tdm_load_2d(unsigned lds_off, const void* gaddr,
                                            int tile0, int tile1, int stride0,
                                            unsigned padi, unsigned pada) {
  unsigned long long ga = (unsigned long long)(uintptr_t)gaddr;
  u32x4 g0;
  g0[0] = 1u;                                       // count=1, user load D#
  g0[1] = lds_off;                                  // LDS byte address
  g0[2] = (unsigned)(ga & 0xffffffffu);             // global_addr[31:0]
  g0[3] = (unsigned)((ga >> 32) & 0x01ffffffu)      // global_addr[56:32]
        | (2u << 30);                               // type = 2 (image/tensor)
  i32x8 g1;
  g1[0] = (int)((2u << 16)      // data_size = 4B
              | (1u << 20)      // pad_enable
              | (padi << 22)    // pad_interval code
              | (pada << 25));  // pad_amount code
  g1[1] = (int)(((unsigned)tile0 & 0xffffu) << 16); // tensor_dim0[15:0]
  g1[2] = (int)(((unsigned)tile1 & 0xffffu) << 16); // tensor_dim1[15:0]
  g1[3] = (int)(((unsigned)tile0 & 0xffffu) << 16); // tile_dim0
  g1[4] = (int)((unsigned)tile1 & 0xffffu);         // tile_dim1 (tile_dim2=0)
  g1[5] = stride0;                                  // tensor_dim0_stride[31:0]
  g1[6] = 0;
  g1[7] = 0;
  i32x4 z4 = {0, 0, 0, 0};                          // groups 2/3 unused (2D)
#if defined(__clang_major__) && (__clang_major__ >= 23)
  i32x8 z8 = {0, 0, 0, 0, 0, 0, 0, 0};
  __builtin_amdgcn_tensor_load_to_lds(g0, g1, z4, z4, z8, 0);
#else
  __builtin_amdgcn_tensor_load_to_lds(g0, g1, z4, z4, 0);
#endif
}
#endif

// ---------------------------------------------------------------------------
// Shared-LDS radix-2 complex FFT, 2048 points, 256 threads.
// sign = -1 forward, +1 inverse (caller scales by 1/NFFT).
// ---------------------------------------------------------------------------
__device__ __forceinline__ void fft2048(float* re, float* im, int tid, float sign) {
  for (int i = tid; i < NFFT; i += 256) {
    int j = (int)(__brev((unsigned)i) >> 21);
    if (j > i) {
      float tr = re[i]; re[i] = re[j]; re[j] = tr;
      float ti = im[i]; im[i] = im[j]; im[j] = ti;
    }
  }
  __syncthreads();
  for (int s = 1; s <= 11; ++s) {
    int m = 1 << s, half = m >> 1;
    for (int k = tid; k < NFFT / 2; k += 256) {
      int pos = k & (half - 1);
      int i0  = ((k >> (s - 1)) << s) + pos;
      int i1  = i0 + half;
      float ang = sign * 6.28318530717958648f * (float)pos / (float)m;
      float wi, wr;
      __sincosf(ang, &wi, &wr);
      float xr = re[i1], xi = im[i1];
      float tr = wr * xr - wi * xi;
      float ti = wr * xi + wi * xr;
      float ur = re[i0], ui = im[i0];
      re[i0] = ur + tr; im[i0] = ui + ti;
      re[i1] = ur - tr; im[i1] = ui - ti;
    }
    __syncthreads();
  }
}

// ---------------------------------------------------------------------------
// Time-embedding MLP: tv[b,h] = silu(silu(sinemb(t) @ Wt1 + bt1) @ Wt2 + bt2)
// ---------------------------------------------------------------------------
__global__ __launch_bounds__(256) void k_temb(const float* __restrict__ t,
                                              const float* __restrict__ Wt1,
                                              const float* __restrict__ bt1,
                                              const float* __restrict__ Wt2,
                                              const float* __restrict__ bt2,
                                              float* __restrict__ tv) {
  __shared__ float emb[TEMB];
  __shared__ float h1[Hn];
  int tid = threadIdx.x;
  const float lfac = -9.21034037197618f / 63.0f; // -ln(10000)/(half-1)
  for (int b = 0; b < Bn; ++b) {
    if (tid < TEMB) {
      int jj = tid & 63;
      float fr = __expf(lfac * (float)jj);
      float a = t[b] * fr;
      emb[tid] = (tid < 64) ? sinf(a) : cosf(a);
    }
    __syncthreads();
    {
      float s = bt1[tid];
      for (int k = 0; k < TEMB; ++k) s += emb[k] * Wt1[k * Hn + tid];
      h1[tid] = s / (1.0f + expf(-s));
    }
    __syncthreads();
    {
      float s = bt2[tid];
      for (int k = 0; k < Hn; ++k) s += h1[k] * Wt2[k * Hn + tid];
      tv[b * Hn + tid] = s / (1.0f + expf(-s));
    }
    __syncthreads();
  }
}

// ---------------------------------------------------------------------------
// tt[b,h] = sum_k tv[b,k] * tW[k,h] + tb[h]
// ---------------------------------------------------------------------------
__global__ __launch_bounds__(256) void k_tt(const float* __restrict__ tv,
                                            const float* __restrict__ tWb,
                                            const float* __restrict__ tbb,
                                            float* __restrict__ tt) {
  __shared__ float tvb[Hn];
  int b = blockIdx.x, tid = threadIdx.x;
  tvb[tid] = tv[b * Hn + tid];
  __syncthreads();
  float s = tbb[tid];
  for (int k = 0; k < Hn; ++k) s += tvb[k] * tWb[k * Hn + tid];
  tt[b * Hn + tid] = s;
}

// ---------------------------------------------------------------------------
// Input projection (WMMA fp32): x[b,h,l] = relu(input[b,l,:] @ W_in + b_in)[h]
// ---------------------------------------------------------------------------
__global__ __launch_bounds__(256) void k_in_proj(const float* __restrict__ inp,
                                                 const float* __restrict__ W_in,
                                                 const float* __restrict__ b_in,
                                                 float* __restrict__ x) {
  int wid = threadIdx.x >> 5, lane = threadIdx.x & 31;
  int t = blockIdx.x * 8 + wid;            // 16384 wave tiles
  int mt = t & 15, nt = t >> 4;
  int h0 = mt * 16, col0 = nt * 16;
  int hi = lane >> 4, lo = lane & 15;
  v8f acc = {};
  for (int k0 = 0; k0 < DIN; k0 += 4) {
    int k = k0 + 2 * hi;
    v2f av, bv;
    av.x = W_in[k * Hn + h0 + lo];
    av.y = W_in[(k + 1) * Hn + h0 + lo];
    float2 bb = *reinterpret_cast<const float2*>(&inp[(col0 + lo) * DIN + k]);
    bv.x = bb.x; bv.y = bb.y;
    acc = __builtin_amdgcn_wmma_f32_16x16x4_f32(false, av, false, bv,
                                                (short)0, acc, false, false);
  }
  int b = col0 >> 10, l = col0 & (Ln - 1);
#pragma unroll
  for (int v = 0; v < 8; ++v) {
    int h = h0 + v + 8 * hi;
    float d = acc[v] + b_in[h];
    x[((b << 8) + h) * Ln + l + lo] = fmaxf(d, 0.0f);
  }
}

// ---------------------------------------------------------------------------
// z = x + tt (broadcast over L); zn = LayerNorm_H(z) * g + b. Tile of 16 l's.
// ---------------------------------------------------------------------------
__global__ __launch_bounds__(256) void k_ln(const float* __restrict__ x,
                                            const float* __restrict__ tt,
                                            const float* __restrict__ lng,
                                            const float* __restrict__ lnb,
                                            float* __restrict__ z,
                                            float* __restrict__ zn) {
  __shared__ float zt[Hn * 17];
  __shared__ float sred[16][16];
  __shared__ float qred[16][16];
  __shared__ float mus[16], rvs[16];
  int tid = threadIdx.x;
  int b = blockIdx.x >> 6;
  int l0 = (blockIdx.x & 63) * 16;
  for (int idx = tid; idx < Hn * 16; idx += 256) {
    int h = idx >> 4, c = idx & 15;
    float v = x[((b << 8) + h) * Ln + l0 + c] + tt[b * Hn + h];
    z[((b << 8) + h) * Ln + l0 + c] = v;
    zt[h * 17 + c] = v;
  }
  __syncthreads();
  {
    int part = tid >> 4, c = tid & 15;
    float s = 0.0f, q = 0.0f;
    for (int hh = 0; hh < 16; ++hh) {
      float v = zt[(part * 16 + hh) * 17 + c];
      s += v; q += v * v;
    }
    sred[part][c] = s; qred[part][c] = q;
  }
  __syncthreads();
  if (tid < 16) {
    float S = 0.0f, Q = 0.0f;
    for (int p = 0; p < 16; ++p) { S += sred[p][tid]; Q += qred[p][tid]; }
    float mu = S * (1.0f / Hn);
    float var = Q * (1.0f / Hn) - mu * mu;
    mus[tid] = mu;
    rvs[tid] = rsqrtf(var + 1e-5f);
  }
  __syncthreads();
  for (int idx = tid; idx < Hn * 16; idx += 256) {
    int h = idx >> 4, c = idx & 15;
    float v = (zt[h * 17 + c] - mus[c]) * rvs[c] * lng[h] + lnb[h];
    zn[((b << 8) + h) * Ln + l0 + c] = v;
  }
}

// ---------------------------------------------------------------------------
// S4 kernel spectrum: per h, build bidirectional time kernel (len 2L) from
// diagonal ZOH discretization and FFT it -> Kf[h][0..2047] (complex).
// ---------------------------------------------------------------------------
__global__ __launch_bounds__(256) void k_sskernel(const float* __restrict__ log_dt,
                                                  const float* __restrict__ A_log_re,
                                                  const float* __restrict__ A_im,
                                                  const float* __restrict__ C_re,
                                                  const float* __restrict__ C_im,
                                                  float2* __restrict__ Kf) {
  __shared__ float re[NFFT], im[NFFT];
  __shared__ float dtar[Nn], dtai[Nn];
  __shared__ float c0r[Nn], c0i[Nn], c1r[Nn], c1i[Nn];
  int h = blockIdx.x, tid = threadIdx.x;
  if (tid < Nn) {
    int n = tid;
    float dt  = expf(log_dt[h]);
    float Are = -expf(A_log_re[h * Nn + n]);
    float Aim = A_im[h * Nn + n];
    float dr = Are * dt, di = Aim * dt;
    float er = expf(dr), sn, cs;
    __sincosf(di, &sn, &cs);
    float numr = er * cs - 1.0f;    // exp(dtA) - 1
    float numi = er * sn;
    float den = 1.0f / (Are * Are + Aim * Aim);
    float gr = (numr * Are + numi * Aim) * den;   // (exp(dtA)-1)/A
    float gi = (numi * Are - numr * Aim) * den;
    float C0r = C_re[h * Nn + n],             C0i = C_im[h * Nn + n];
    float C1r = C_re[Hn * Nn + h * Nn + n],   C1i = C_im[Hn * Nn + h * Nn + n];
    c0r[n] = C0r * gr - C0i * gi;  c0i[n] = C0r * gi + C0i * gr;
    c1r[n] = C1r * gr - C1i * gi;  c1i[n] = C1r * gi + C1i * gr;
    dtar[n] = dr;  dtai[n] = di;
  }
  __syncthreads();
  for (int l = tid; l < Ln; l += 256) {
    float k0 = 0.0f, k1 = 0.0f;
    float lf = (float)l;
    for (int n = 0; n < Nn; ++n) {
      float wr = __expf(dtar[n] * lf);
      float sn, cs;
      sincosf(dtai[n] * lf, &sn, &cs);
      float vr = wr * cs, vi = wr * sn;
      k0 += c0r[n] * vr - c0i[n] * vi;
      k1 += c1r[n] * vr - c1i[n] * vi;
    }
    re[l] = 2.0f * k0;               im[l] = 0.0f;
    re[2 * Ln - 1 - l] = 2.0f * k1;  im[2 * Ln - 1 - l] = 0.0f;
  }
  __syncthreads();
  fft2048(re, im, tid, -1.0f);
  for (int i = tid; i < NFFT; i += 256) {
    float2 v; v.x = re[i]; v.y = im[i];
    Kf[h * NFFT + i] = v;
  }
}

// ---------------------------------------------------------------------------
// FFT convolution per (b,h) row: y = irfft(fft(pad(zn)) * Kf)[:L] + zn*Dp,
// then exact GELU. Written in place over zn (becomes y).
// ---------------------------------------------------------------------------
__global__ __launch_bounds__(256) void k_fftconv(float* __restrict__ zn,
                                                 const float2* __restrict__ Kf,
                                                 const float* __restrict__ Dp) {
  __shared__ float re[NFFT], im[NFFT];
  int row = blockIdx.x, tid = threadIdx.x;
  int h = row & (Hn - 1);
  long base = (long)row * Ln;
  for (int i = tid; i < NFFT; i += 256) {
    re[i] = (i < Ln) ? zn[base + i] : 0.0f;
    im[i] = 0.0f;
  }
  __syncthreads();
  fft2048(re, im, tid, -1.0f);
  for (int i = tid; i < NFFT; i += 256) {
    float2 kf = Kf[h * NFFT + i];
    float xr = re[i], xi = im[i];
    re[i] = xr * kf.x - xi * kf.y;
    im[i] = xr * kf.y + xi * kf.x;
  }
  __syncthreads();
  fft2048(re, im, tid, 1.0f);
  float dp = Dp[h];
  for (int l = tid; l < Ln; l += 256) {
    float conv = re[l] * (1.0f / NFFT);
    float v = conv + zn[base + l] * dp;
    v = 0.5f * v * (1.0f + erff(v * 0.70710678118654752f));
    zn[base + l] = v;   // y, in place
  }
}

// ---------------------------------------------------------------------------
// Channel GEMM (WMMA fp32): out[o, (b,l)] = sum_h W[o,h] * in[b,h,l] + bias[o]
// A panel (32x256) and double-buffered B chunks (32x64) staged in LDS by the
// Tensor Data Mover (wave 0 issues, TENSORcnt-paced; TDM row-padding gives
// the conflict-free LDS strides 260 / 72 directly). Synchronous staging
// fallback for toolchains without the tensor builtins, and for
// transposed-weight modes (A only).
// Modes: 0 = +add, tanh*sigmoid -> out[B,H,L]   (s4 output + GLU gate)
//        1 = +add (nullable)    -> out[B,H,L]   (residual / skip accumulate)
//        3 = relu               -> out[B,H,L]   (out-proj layer 1)
//        4 = +add[B,L,D]        -> out[B,L,D]   (final projection, M=64)
// ---------------------------------------------------------------------------
__global__ __launch_bounds__(256) void k_cgemm(const float* __restrict__ W,
                                               int wsm, int wsk,
                                               const float* __restrict__ bias,
                                               const float* __restrict__ in,
                                               const float* __restrict__ add,
                                               float* __restrict__ out,
                                               int mode) {
  __shared__ float ldsA[32 * 260];
  __shared__ float ldsB[2][KCH * BROW];
  int tid = threadIdx.x;
  int m0    = (blockIdx.x >> 8) * 32;
  int ncol0 = (blockIdx.x & 255) * 64;
  int wid = tid >> 5, lane = tid & 31;
  int waveM = wid >> 2, waveN = wid & 3;
  int hi = lane >> 4, lo = lane & 15;
  int b = ncol0 >> 10, l0 = ncol0 & (Ln - 1);
  const float* inBase = in + ((long)(b << 8)) * Ln + l0;  // row k at +k*Ln

#if USE_TDM
  // A panel: row-major weights -> one 2D TDM (rows of 256 fp32, stride wsm;
  // pad 4 DWORDs per 256-DWORD row -> LDS row stride 260). Transposed
  // weights (wsk!=1) use the cooperative strided copy.
  if (wsk == 1) {
    if (wid == 0)
      tdm_load_2d((unsigned)(uintptr_t)&ldsA[0], W + (long)m0 * wsm,
                  Hn, 32, wsm, 7u, 3u);
  } else {
    for (int idx = tid; idx < 32 * Hn; idx += 256) {
      int ml = idx >> 8, k = idx & 255;
      ldsA[ml * 260 + k] = W[(m0 + ml) * wsm + k * wsk];
    }
  }
  // B chunk 0: 32 rows x 64 fp32, global row stride Ln; pad 8 DWORDs per
  // 64-DWORD row -> LDS row stride 72.
  if (wid == 0) {
    tdm_load_2d((unsigned)(uintptr_t)&ldsB[0][0], inBase, 64, KCH, Ln, 5u, 7u);
  }
#else
  for (int idx = tid; idx < 32 * Hn; idx += 256) {
    int ml = idx >> 8, k = idx & 255;
    ldsA[ml * 260 + k] = W[(m0 + ml) * wsm + k * wsk];
  }
  for (int idx = tid; idx < KCH * 64; idx += 256) {
    int kk = idx >> 6, c = idx & 63;
    ldsB[0][kk * BROW + c] = inBase[(long)kk * Ln + c];
  }
#endif

  const float* arow = &ldsA[(waveM * 16 + lo) * 260];
  int ncolw = waveN * 16 + lo;
  v8f acc = {};
  for (int kc = 0; kc < NCH; ++kc) {
#if USE_TDM
    if (wid == 0) {
      if (kc + 1 < NCH) {
        tdm_load_2d((unsigned)(uintptr_t)&ldsB[(kc + 1) & 1][0],
                    inBase + (long)(kc + 1) * KCH * Ln, 64, KCH, Ln, 5u, 7u);
        // In-order TENSORcnt: <=1 outstanding drains A + B[kc]; B[kc+1] flies.
        __builtin_amdgcn_s_wait_tensorcnt(1);
      } else {
        __builtin_amdgcn_s_wait_tensorcnt(0);
      }
    }
#endif
    __syncthreads();   // chunk kc resident in LDS for all waves
    const float* bbuf = &ldsB[kc & 1][0];
#pragma unroll
    for (int kk = 0; kk < KCH; kk += 4) {
      int k2 = kk + 2 * hi;
      float2 a2 = *reinterpret_cast<const float2*>(&arow[kc * KCH + k2]);
      v2f av; av.x = a2.x; av.y = a2.y;
      v2f bv;
      bv.x = bbuf[k2 * BROW + ncolw];
      bv.y = bbuf[(k2 + 1) * BROW + ncolw];
      acc = __builtin_amdgcn_wmma_f32_16x16x4_f32(false, av, false, bv,
                                                  (short)0, acc, false, false);
    }
    __syncthreads();   // all waves done reading buf[kc&1] before it is reused
#if !USE_TDM
    if (kc + 1 < NCH) {
      for (int idx = tid; idx < KCH * 64; idx += 256) {
        int kk = idx >> 6, c = idx & 63;
        ldsB[(kc + 1) & 1][kk * BROW + c] =
            inBase[(long)((kc + 1) * KCH + kk) * Ln + c];
      }
    }
#endif
  }

#pragma unroll
  for (int v = 0; v < 8; ++v) {
    int o = m0 + waveM * 16 + v + 8 * hi;
    float d = acc[v] + bias[o];
    if (mode == 0) {
      long idx = ((long)((b << 8) + o)) * Ln + l0 + waveN * 16 + lo;
      float val = d + add[idx];
      out[idx] = tanhf(val) * (1.0f / (1.0f + expf(-val)));
    } else if (mode == 1) {
      long idx = ((long)((b << 8) + o)) * Ln + l0 + waveN * 16 + lo;
      float val = d + (add ? add[idx] : 0.0f);
      out[idx] = val;
    } else if (mode == 3) {
      long idx = ((long)((b << 8) + o)) * Ln + l0 + waveN * 16 + lo;
      out[idx] = fmaxf(d, 0.0f);
    } else { // mode 4: final [B,L,DIN] layout, residual with input
      long idx = ((long)((b << 10) + l0 + waveN * 16 + lo)) * DIN + o;
      out[idx] = d + add[idx];
    }
  }
}

// ---------------------------------------------------------------------------
// Host-side launch
// ---------------------------------------------------------------------------
extern "C" void kernel_launch(void* const* d_in, const int* in_sizes, int n_in,
                              void* d_out, int out_size, void* d_ws, size_t ws_size,
                              hipStream_t stream) {
  const float* input    = (const float*)d_in[0];
  const float* t        = (const float*)d_in[1];
  const float* W_in     = (const float*)d_in[2];
  const float* b_in     = (const float*)d_in[3];
  const float* Wt1      = (const float*)d_in[4];
  const float* bt1      = (const float*)d_in[5];
  const float* Wt2      = (const float*)d_in[6];
  const float* bt2      = (const float*)d_in[7];
  const float* ln_g     = (const float*)d_in[8];
  const float* ln_b     = (const float*)d_in[9];
  const float* log_dt   = (const float*)d_in[10];
  const float* A_log_re = (const float*)d_in[11];
  const float* A_im     = (const float*)d_in[12];
  const float* C_re     = (const float*)d_in[13];
  const float* C_im     = (const float*)d_in[14];
  const float* Dp       = (const float*)d_in[15];
  const float* s4W      = (const float*)d_in[16];
  const float* s4b      = (const float*)d_in[17];
  const float* tW       = (const float*)d_in[18];
  const float* tb       = (const float*)d_in[19];
  const float* c1W      = (const float*)d_in[20];
  const float* c1b      = (const float*)d_in[21];
  const float* c2W      = (const float*)d_in[22];
  const float* c2b      = (const float*)d_in[23];
  const float* Wo1      = (const float*)d_in[24];
  const float* bo1      = (const float*)d_in[25];
  const float* Wo2      = (const float*)d_in[26];
  const float* bo2      = (const float*)d_in[27];

  float* ws = (float*)d_ws;
  const size_t T = (size_t)Bn * Hn * Ln;   // 4,194,304 floats / tensor
  float* x     = ws;
  float* z     = ws + 1 * T;
  float* zn    = ws + 2 * T;               // becomes y after fftconv
  float* g     = ws + 3 * T;               // reused as h1 at the end
  float* skips = ws + 4 * T;
  float* Kf    = ws + 5 * T;               // 256*2048 complex = 1,048,576 floats
  float* tv    = ws + 5 * T + (size_t)Hn * NFFT * 2;
  float* tt    = tv + Bn * Hn;

  k_temb<<<1, 256, 0, stream>>>(t, Wt1, bt1, Wt2, bt2, tv);
  k_in_proj<<<2048, 256, 0, stream>>>(input, W_in, b_in, x);

  for (int ib = 0; ib < NBLK; ++ib) {
    const int oH = ib * Hn, oHH = ib * Hn * Hn, oHN = ib * Hn * Nn;
    k_tt<<<Bn, 256, 0, stream>>>(tv, tW + oHH, tb + oH, tt);
    k_sskernel<<<Hn, 256, 0, stream>>>(log_dt + oH, A_log_re + oHN, A_im + oHN,
                                       C_re + ib * 2 * Hn * Nn,
                                       C_im + ib * 2 * Hn * Nn,
                                       (float2*)Kf);
    k_ln<<<Bn * (Ln / 16), 256, 0, stream>>>(x, tt, ln_g + oH, ln_b + oH, z, zn);
    k_fftconv<<<Bn * Hn, 256, 0, stream>>>(zn, (const float2*)Kf, Dp + oH);
    // y(=zn) @ s4W + s4b + z -> tanh*sigmoid gate -> g
    k_cgemm<<<2048, 256, 0, stream>>>(s4W + oHH, Hn, 1, s4b + oH, zn, z, g, 0);
    // x = c1W @ g + c1b + x   (in place, element read-once/write-once)
    k_cgemm<<<2048, 256, 0, stream>>>(c1W + oHH, Hn, 1, c1b + oH, g, x, x, 1);
    // skips = c2W @ g + c2b (+ skips)  -- first block initializes
    k_cgemm<<<2048, 256, 0, stream>>>(c2W + oHH, Hn, 1, c2b + oH, g,
                                      (ib == 0) ? nullptr : skips, skips, 1);
  }

  // out-proj: h1 = relu(skips^T @ Wo1 + bo1)  (Wo1 read transposed)
  k_cgemm<<<2048, 256, 0, stream>>>(Wo1, 1, Hn, bo1, skips, nullptr, g, 3);
  // out = h1 @ Wo2 + bo2 + input, stored [B,L,DIN]
  k_cgemm<<<512, 256, 0, stream>>>(Wo2, 1, DIN, bo2, g, input, (float*)d_out, 4);
}